// GATConv_34986803593623
// MI455X (gfx1250) — compile-verified
//
#include <hip/hip_runtime.h>

#define NODES 100000
#define EDGES 1600000
#define DIN   256
#define DEDGE 16
#define HEADS 4
#define FEAT  32
#define HF    128        // HEADS*FEAT
#define NEG_SLOPE 0.2f

typedef float v2f __attribute__((ext_vector_type(2)));
typedef float v8f __attribute__((ext_vector_type(8)));

// ---------------------------------------------------------------------------
// Kernel 1: feat_src = X@W_src, feat_dst = X@W_dst (written to out),
//           attn_src = X@A_src, attn_dst = X@A_dst.
// One block = 16 node rows. 8 waves, each owns one 16-wide column tile and
// accumulates both W_src and W_dst products with V_WMMA_F32_16X16X4_F32.
// ---------------------------------------------------------------------------
__global__ __launch_bounds__(256) void gat_node_gemm(
    const float* __restrict__ X,
    const float* __restrict__ Wsrc, const float* __restrict__ Wdst,
    const float* __restrict__ Asrc, const float* __restrict__ Adst,
    float* __restrict__ feat_src, float* __restrict__ out,
    float* __restrict__ attn_src, float* __restrict__ attn_dst)
{
    // Row padded 256 -> 260 floats: A-fragment column reads hit distinct banks.
    __shared__ float xs[16][DIN + 4];

    const int row0 = blockIdx.x * 16;
    const int tid  = threadIdx.x;

    // Cooperative, coalesced load of the 16x256 X tile.
    for (int i = tid; i < 16 * DIN; i += 256) {
        const int r = i >> 8;       // DIN == 256
        const int c = i & 255;
        xs[r][c] = X[(size_t)(row0 + r) * DIN + c];
    }
    __syncthreads();

    const int wave = tid >> 5;
    const int lane = tid & 31;
    const int nl   = lane & 15;            // M for A-frag, N for B-frag
    const int kb   = (lane >> 4) << 1;     // K sub-offset: half-wave split
    const int col0 = wave * 16;

    v8f acc_s = {};
    v8f acc_d = {};

    for (int k = 0; k < DIN; k += 4) {
        const int kk = k + kb;
        // A-fragment: 16x4 f32, lane = row M, VGPR pair = (K, K+1).
        v2f a;
        a.x = xs[nl][kk];
        a.y = xs[nl][kk + 1];
        // B-fragments: 4x16 f32, lane = col N, VGPR pair = rows (K, K+1).
        v2f bs, bd;
        const size_t wo0 = (size_t)kk * HF + col0 + nl;
        const size_t wo1 = (size_t)(kk + 1) * HF + col0 + nl;
        bs.x = Wsrc[wo0];  bs.y = Wsrc[wo1];
        bd.x = Wdst[wo0];  bd.y = Wdst[wo1];

        acc_s = __builtin_amdgcn_wmma_f32_16x16x4_f32(
            false, a, false, bs, (short)0, acc_s, false, false);
        acc_d = __builtin_amdgcn_wmma_f32_16x16x4_f32(
            false, a, false, bd, (short)0, acc_d, false, false);
    }

    // C/D layout: VGPR r, lanes 0-15 -> row r; lanes 16-31 -> row r+8.
    const int rbase = (lane >> 4) * 8;
#pragma unroll
    for (int r = 0; r < 8; ++r) {
        const size_t o = (size_t)(row0 + rbase + r) * HF + col0 + nl;
        feat_src[o] = acc_s[r];
        out[o]      = acc_d[r];   // residual feat_dst pre-seeded into output
    }

    // Attention projections from the LDS-resident tile.
    // 128 threads: r in [0,16), h in [0,4), which selects A_src/A_dst.
    if (tid < 128) {
        const int r     = tid >> 3;
        const int h     = tid & 3;
        const int which = (tid >> 2) & 1;
        const float* A  = which ? Adst : Asrc;
        float acc = 0.f;
#pragma unroll 8
        for (int k = 0; k < DIN; ++k) acc += xs[r][k] * A[k * HEADS + h];
        float* o = which ? attn_dst : attn_src;
        o[(size_t)(row0 + r) * HEADS + h] = acc;
    }
}

// ---------------------------------------------------------------------------
// Kernel 2: init segment max (-inf) and segment sum (0).
// ---------------------------------------------------------------------------
__global__ void gat_init(float* __restrict__ m, float* __restrict__ z)
{
    const int i = blockIdx.x * blockDim.x + threadIdx.x;
    if (i < NODES * HEADS) {
        m[i] = -__builtin_inff();
        z[i] = 0.f;
    }
}

// ---------------------------------------------------------------------------
// Kernel 3: per-edge logits + leaky-ReLU + atomic segment max over dst.
// ---------------------------------------------------------------------------
__global__ __launch_bounds__(256) void gat_edge_logits(
    const float* __restrict__ edge_in,
    const int* __restrict__ src, const int* __restrict__ dst,
    const float* __restrict__ Aedge,
    const float* __restrict__ attn_src, const float* __restrict__ attn_dst,
    float* __restrict__ ebuf, float* __restrict__ m)
{
    const int e = blockIdx.x * blockDim.x + threadIdx.x;
    if (e >= EDGES) return;

    float ae[HEADS] = {0.f, 0.f, 0.f, 0.f};
    const float* er = edge_in + (size_t)e * DEDGE;
#pragma unroll
    for (int k = 0; k < DEDGE; ++k) {
        const float x = er[k];
#pragma unroll
        for (int h = 0; h < HEADS; ++h) ae[h] += x * Aedge[k * HEADS + h];
    }

    const int s = src[e];
    const int d = dst[e];
#pragma unroll
    for (int h = 0; h < HEADS; ++h) {
        float v = attn_src[(size_t)s * HEADS + h]
                + attn_dst[(size_t)d * HEADS + h] + ae[h];
        v = v > 0.f ? v : NEG_SLOPE * v;
        ebuf[(size_t)e * HEADS + h] = v;
        atomicMax(&m[(size_t)d * HEADS + h], v);
    }
}

// ---------------------------------------------------------------------------
// Kernel 4: exp(e - m[dst]) + atomic segment sum into z.
// ---------------------------------------------------------------------------
__global__ __launch_bounds__(256) void gat_edge_exp(
    const int* __restrict__ dst, const float* __restrict__ m,
    float* __restrict__ ebuf, float* __restrict__ z)
{
    const int e = blockIdx.x * blockDim.x + threadIdx.x;
    if (e >= EDGES) return;
    const int d = dst[e];
#pragma unroll
    for (int h = 0; h < HEADS; ++h) {
        const float w = __expf(ebuf[(size_t)e * HEADS + h]
                               - m[(size_t)d * HEADS + h]);
        ebuf[(size_t)e * HEADS + h] = w;
        atomicAdd(&z[(size_t)d * HEADS + h], w);
    }
}

// ---------------------------------------------------------------------------
// Kernel 5: alpha-weighted gather/scatter. One wave per edge; lane = feature.
// ---------------------------------------------------------------------------
__global__ __launch_bounds__(256) void gat_scatter(
    const int* __restrict__ src, const int* __restrict__ dst,
    const float* __restrict__ feat_src, const float* __restrict__ ebuf,
    const float* __restrict__ z, float* __restrict__ out)
{
    const int e = blockIdx.x * 8 + (threadIdx.x >> 5);
    const int lane = threadIdx.x & 31;
    if (e >= EDGES) return;

    const int s = src[e];
    const int d = dst[e];
#pragma unroll
    for (int h = 0; h < HEADS; ++h) {
        const float zz    = z[(size_t)d * HEADS + h];
        const float alpha = ebuf[(size_t)e * HEADS + h] / fmaxf(zz, 1e-16f);
        const int   f     = h * FEAT + lane;
        const float msg   = feat_src[(size_t)s * HF + f] * alpha;
        atomicAdd(&out[(size_t)d * HF + f], msg);
    }
}

// ---------------------------------------------------------------------------
extern "C" void kernel_launch(void* const* d_in, const int* in_sizes, int n_in,
                              void* d_out, int out_size, void* d_ws, size_t ws_size,
                              hipStream_t stream)
{
    const float* X     = (const float*)d_in[0];
    const float* Ein   = (const float*)d_in[1];
    const int*   src   = (const int*)  d_in[2];
    const int*   dst   = (const int*)  d_in[3];
    const float* Wsrc  = (const float*)d_in[4];
    const float* Wdst  = (const float*)d_in[5];
    const float* Asrc  = (const float*)d_in[6];
    const float* Adst  = (const float*)d_in[7];
    const float* Aedge = (const float*)d_in[8];
    float* out = (float*)d_out;

    float* feat_src = (float*)d_ws;                              // N*128
    float* attn_src = feat_src + (size_t)NODES * HF;             // N*4
    float* attn_dst = attn_src + (size_t)NODES * HEADS;          // N*4
    float* mbuf     = attn_dst + (size_t)NODES * HEADS;          // N*4
    float* zbuf     = mbuf     + (size_t)NODES * HEADS;          // N*4
    float* ebuf     = zbuf     + (size_t)NODES * HEADS;          // E*4

    gat_node_gemm<<<NODES / 16, 256, 0, stream>>>(
        X, Wsrc, Wdst, Asrc, Adst, feat_src, out, attn_src, attn_dst);

    gat_init<<<(NODES * HEADS + 255) / 256, 256, 0, stream>>>(mbuf, zbuf);

    gat_edge_logits<<<(EDGES + 255) / 256, 256, 0, stream>>>(
        Ein, src, dst, Aedge, attn_src, attn_dst, ebuf, mbuf);

    gat_edge_exp<<<(EDGES + 255) / 256, 256, 0, stream>>>(dst, mbuf, ebuf, zbuf);

    gat_scatter<<<EDGES / 8, 256, 0, stream>>>(
        src, dst, feat_src, ebuf, zbuf, out);
}